// EdgeConvolution_67319317397679
// MI455X (gfx1250) — compile-verified
//
#include <hip/hip_runtime.h>
#include <stdint.h>

typedef __attribute__((ext_vector_type(2))) float v2f;
typedef __attribute__((ext_vector_type(8))) float v8f;

// Problem constants (match the reference)
constexpr int Bc = 32, Nc = 192, Dc = 128, Oc = 128, Ec = 4;
constexpr int NT = Nc / 16;  // 12 tiles over N
constexpr int OT = Oc / 16;  // 8 tiles over O
constexpr int LDS_PITCH = 132;  // 128 + 4 floats: de-conflicts the two lane halves

// ---------------------------------------------------------------------------
// Stage 1: A[b,e,n,o] = nodes[b,n,:] @ W1[e,:,o]
//          C[b,e,n,o] = nodes[b,n,:] @ W2[e,:,o] + bias[e,o]
// One wave per (b,e, 16-row n-tile, 16-col o-tile); K-loop over D in steps
// of 4 using V_WMMA_F32_16X16X4_F32.
// ---------------------------------------------------------------------------
__global__ __launch_bounds__(32)
void stage1_gemm(const float* __restrict__ nodes,
                 const float* __restrict__ W,
                 const float* __restrict__ bias,
                 float* __restrict__ A_ws,
                 float* __restrict__ C_ws) {
  int bid = blockIdx.x;
  int ob = bid % OT; bid /= OT;
  int it = bid % NT; bid /= NT;
  int e  = bid % Ec; bid /= Ec;
  int b  = bid;

  const int lane = threadIdx.x;
  const int hi = lane >> 4;   // lane half selects K pair {0,1} vs {2,3}
  const int ln = lane & 15;

  const float* nodeRow = nodes + ((size_t)b * Nc + it * 16 + ln) * Dc;
  const float* W1 = W + (size_t)e * (2 * Dc) * Oc;   // W[e, 0:D,  :]
  const float* W2 = W1 + (size_t)Dc * Oc;            // W[e, D:2D, :]

  v8f accA = {};
  v8f accC = {};

  for (int k0 = 0; k0 < Dc; k0 += 4) {
    const int ka = k0 + 2 * hi;                 // K for VGPR0 of this lane
    v2f a = *(const v2f*)(nodeRow + ka);        // A 16x4: VGPR0={K0|K2}, VGPR1={K1|K3}

    const float* w1c = W1 + (size_t)ka * Oc + ob * 16 + ln;
    const float* w2c = W2 + (size_t)ka * Oc + ob * 16 + ln;
    v2f b1; b1.x = w1c[0]; b1.y = w1c[Oc];
    v2f b2; b2.x = w2c[0]; b2.y = w2c[Oc];

    accA = __builtin_amdgcn_wmma_f32_16x16x4_f32(false, a, false, b1,
                                                 (short)0, accA, false, false);
    accC = __builtin_amdgcn_wmma_f32_16x16x4_f32(false, a, false, b2,
                                                 (short)0, accC, false, false);
  }

  const float bb = bias[e * Oc + ob * 16 + ln];

  size_t base = (((size_t)b * Ec + e) * Nc + it * 16 + 8 * hi) * Oc + ob * 16 + ln;
#pragma unroll
  for (int v = 0; v < 8; ++v) {
    A_ws[base + (size_t)v * Oc] = accA[v];
    C_ws[base + (size_t)v * Oc] = accC[v] + bb;
  }
}

// ---------------------------------------------------------------------------
// Stage 2 (single fused pass over the 604 MB output):
//   out[b,i,j,o] = sum_e adj[b,e,i,j] * A[b,e,i,o]   (K=4 WMMA, M=j, N=o)
//                + sum_e adj[b,e,i,j] * C[b,e,j,o]   (diagonal in j: VALU+LDS)
// Block = 8 waves sharing (b, j-tile), wave w handles i = ig*8 + w.
// The shared 32 KB C[b,:,j-tile,:] slab is staged in LDS once per block via
// GLOBAL_LOAD_ASYNC_TO_LDS_B128 (ASYNCcnt), cutting L2 re-reads 8x.
// Output uses non-temporal stores so the 604 MB stream does not evict the
// L2-resident workspace.
// ---------------------------------------------------------------------------
__global__ __launch_bounds__(256)
void stage2_fused(const float* __restrict__ adj,
                  const float* __restrict__ A_ws,
                  const float* __restrict__ C_ws,
                  float* __restrict__ out) {
  const int jt = blockIdx.x;        // 0..NT-1
  const int ig = blockIdx.y;        // 0..N/8-1
  const int b  = blockIdx.z;        // 0..B-1

  const int tid  = threadIdx.x;
  const int wave = tid >> 5;
  const int lane = tid & 31;
  const int hi = lane >> 4;
  const int ln = lane & 15;
  const int i  = ig * 8 + wave;

  __shared__ float ldsC[Ec][16][LDS_PITCH];   // 33 KB, rows padded to 528 B

  // ---- cooperative async copy: C_ws[b, 0:4, jt*16:+16, 0:128] -> LDS ----
  {
    const uint32_t ldsBase = (uint32_t)(uintptr_t)(&ldsC[0][0][0]);
#pragma unroll
    for (int r = 0; r < 8; ++r) {
      const int flat = tid + r * 256;     // 0..2047 b128 transfers
      const int row  = flat >> 5;         // e*16 + j_local, 0..63
      const int c16  = flat & 31;         // 16 B column within the 512 B row
      const float* g = C_ws +
          ((((size_t)b * Ec + (row >> 4)) * Nc) + jt * 16 + (row & 15)) * Oc + c16 * 4;
      const uint32_t ldsAddr = ldsBase + (uint32_t)(row * (LDS_PITCH * 4) + c16 * 16);
      const uint64_t ga = (uint64_t)(uintptr_t)g;
      asm volatile("global_load_async_to_lds_b128 %0, %1, off"
                   :: "v"(ldsAddr), "v"(ga) : "memory");
    }
    asm volatile("s_wait_asynccnt 0x0" ::: "memory");
  }
  __syncthreads();

  const size_t eStrideAdj = (size_t)Nc * Nc;  // adj e-stride
  const size_t eStrideWS  = (size_t)Nc * Oc;  // A_ws e-stride

  // adj fragment: M = j = ln, K = e (VGPR0 = {e0|e2}, VGPR1 = {e1|e3})
  const size_t adjBase = (((size_t)b * Ec) * Nc + i) * Nc;  // + e*N*N + j
  v2f af;
  af.x = adj[adjBase + (size_t)(2 * hi)     * eStrideAdj + jt * 16 + ln];
  af.y = adj[adjBase + (size_t)(2 * hi + 1) * eStrideAdj + jt * 16 + ln];

  v8f acc[OT];
#pragma unroll
  for (int ob = 0; ob < OT; ++ob) acc[ob] = (v8f){};

  // ---- term1: 8 x V_WMMA_F32_16X16X4_F32 ----
  const size_t aBase = (((size_t)b * Ec + 2 * hi) * Nc + i) * Oc + ln;
#pragma unroll
  for (int ob = 0; ob < OT; ++ob) {
    v2f bf;
    bf.x = A_ws[aBase + ob * 16];
    bf.y = A_ws[aBase + ob * 16 + eStrideWS];
    acc[ob] = __builtin_amdgcn_wmma_f32_16x16x4_f32(false, af, false, bf,
                                                    (short)0, acc[ob], false, false);
  }

  // ---- term2: diagonal in j; C operands from LDS, fused VALU FMAs ----
#pragma unroll
  for (int v = 0; v < 8; ++v) {
    const int jl = v + 8 * hi;                 // local j = accumulator row M
    const int j  = jt * 16 + jl;
    const float a0 = adj[adjBase + 0 * eStrideAdj + j];
    const float a1 = adj[adjBase + 1 * eStrideAdj + j];
    const float a2 = adj[adjBase + 2 * eStrideAdj + j];
    const float a3 = adj[adjBase + 3 * eStrideAdj + j];
#pragma unroll
    for (int ob = 0; ob < OT; ++ob) {
      const int o = ob * 16 + ln;
      float s = a0 * ldsC[0][jl][o];
      s = fmaf(a1, ldsC[1][jl][o], s);
      s = fmaf(a2, ldsC[2][jl][o], s);
      s = fmaf(a3, ldsC[3][jl][o], s);
      acc[ob][v] += s;
    }
  }

  // ---- store: 16 contiguous rows of 512 B per wave, non-temporal ----
#pragma unroll
  for (int v = 0; v < 8; ++v) {
    const size_t o0 = (((size_t)b * Nc + i) * Nc + jt * 16 + v + 8 * hi) * Oc + ln;
#pragma unroll
    for (int ob = 0; ob < OT; ++ob)
      __builtin_nontemporal_store(acc[ob][v], &out[o0 + ob * 16]);
  }
}

// ---------------------------------------------------------------------------
// Harness entry point
// ---------------------------------------------------------------------------
extern "C" void kernel_launch(void* const* d_in, const int* in_sizes, int n_in,
                              void* d_out, int out_size, void* d_ws, size_t ws_size,
                              hipStream_t stream) {
  const float* nodes = (const float*)d_in[0];  // (B, N, D)
  const float* adj   = (const float*)d_in[1];  // (B, E, N, N)
  const float* W     = (const float*)d_in[2];  // (E, 2D, O)
  const float* bias  = (const float*)d_in[3];  // (E, O)
  float* out = (float*)d_out;                  // (B, N, N, O)

  float* A_ws = (float*)d_ws;                          // B*E*N*O floats (12.6 MB)
  float* C_ws = A_ws + (size_t)Bc * Ec * Nc * Oc;      // B*E*N*O floats (12.6 MB)

  stage1_gemm<<<Bc * Ec * NT * OT, 32, 0, stream>>>(nodes, W, bias, A_ws, C_ws);

  dim3 grid2(NT, Nc / 8, Bc);
  stage2_fused<<<grid2, 256, 0, stream>>>(adj, A_ws, C_ws, out);
}